// KeypointLoss_64879775973882
// MI455X (gfx1250) — compile-verified
//
#include <hip/hip_runtime.h>

typedef __attribute__((ext_vector_type(2))) float v2f;
typedef __attribute__((ext_vector_type(8))) float v8f;

#define B_   8
#define N_   16
#define K_   128
#define H_   64
#define W_   64
#define HW_  4096
#define CAP  2048
#define THREADS 256

// One workgroup per (b,n) heatmap. Phases:
//  1) compact positive pixels into LDS as the 4-vector [-2r, -2c, 1, r^2+c^2]
//  2) per wave: WMMA f32 16x16x4  D = A(16 keypoints x 4) * B(4 x 16 pixels)
//     giving squared distances; running min over pixel tiles; shfl_xor reduce
//  3) score loss terms + deterministic LDS tree reduction -> per-block partials
__global__ __launch_bounds__(THREADS) void keypoint_loss_main(
    const float* __restrict__ scores,    // [B,K,N]
    const float* __restrict__ heat,      // [B,N,H,W]
    const float* __restrict__ kpts,      // [B,K,2]
    float* __restrict__ partials)        // [B*N, 4] (pos, neg, cnt, pad)
{
    __shared__ float4   sB[CAP];
    __shared__ float    sDmin[K_];
    __shared__ float    sR0[128];
    __shared__ float    sR1[128];
    __shared__ float    sR2[128];
    __shared__ unsigned sCount;

    const int tid = threadIdx.x;
    const int bn  = blockIdx.x;        // 0..127
    const int b   = bn >> 4;
    const int n   = bn & (N_ - 1);

    if (tid == 0) sCount = 0u;
    __syncthreads();

    // ---- Phase 1: compact positive pixels (coalesced scan) ----
    const float* hrow = heat + (size_t)bn * HW_;
    #pragma unroll
    for (int i = 0; i < HW_ / THREADS; ++i) {
        int p = i * THREADS + tid;
        if (hrow[p] > 0.0f) {
            unsigned idx = atomicAdd(&sCount, 1u);   // ds_add_rtn_u32
            if (idx < CAP) {
                float r = (float)(p >> 6);
                float c = (float)(p & 63);
                sB[idx] = make_float4(-2.0f * r, -2.0f * c, 1.0f, r * r + c * c);
            }
        }
    }
    __syncthreads();

    unsigned count = sCount;
    if (count > CAP) count = CAP;
    unsigned padded = (count + 15u) & ~15u;          // >= 16 (pixel (0,0) always set)
    for (unsigned i = count + (unsigned)tid; i < padded; i += THREADS)
        sB[i] = make_float4(0.0f, 0.0f, 1.0f, 1.0e30f);   // sentinel: huge d^2
    __syncthreads();

    // ---- Phase 2: WMMA distance tiles ----
    const int lane  = tid & 31;
    const int wave  = tid >> 5;        // 8 waves -> 8 M-tiles of 16 keypoints
    const int kbase = wave * 16;
    const int r16   = lane & 15;

    // A-matrix 16x4 f32 layout: lanes 0-15 hold K=0,1 ; lanes 16-31 hold K=2,3
    float kx = kpts[((size_t)b * K_ + kbase + r16) * 2 + 0];
    float ky = kpts[((size_t)b * K_ + kbase + r16) * 2 + 1];
    v2f a;
    a.x = (lane < 16) ? kx : (kx * kx + ky * ky);
    a.y = (lane < 16) ? ky : 1.0f;

    v8f mind;
    #pragma unroll
    for (int r = 0; r < 8; ++r) mind[r] = 3.0e38f;

    const int ntiles = (int)(padded >> 4);
    for (int t = 0; t < ntiles; ++t) {
        float4 bv = sB[t * 16 + r16];                 // ds_load_b128
        v2f bb;
        bb.x = (lane < 16) ? bv.x : bv.z;             // K=0 / K=2 rows
        bb.y = (lane < 16) ? bv.y : bv.w;             // K=1 / K=3 rows
        v8f c = {};
        v8f d = __builtin_amdgcn_wmma_f32_16x16x4_f32(
            false, a, false, bb, (short)0, c, false, false);
        #pragma unroll
        for (int r = 0; r < 8; ++r) mind[r] = fminf(mind[r], d[r]);
    }

    // Min across the 16 lanes holding each row (halves reduce independently)
    #pragma unroll
    for (int off = 8; off >= 1; off >>= 1) {
        #pragma unroll
        for (int r = 0; r < 8; ++r)
            mind[r] = fminf(mind[r], __shfl_xor(mind[r], off, 32));
    }
    if (lane == 0) {
        #pragma unroll
        for (int r = 0; r < 8; ++r) sDmin[kbase + r] = mind[r];       // M = r
    }
    if (lane == 16) {
        #pragma unroll
        for (int r = 0; r < 8; ++r) sDmin[kbase + 8 + r] = mind[r];   // M = r+8
    }
    __syncthreads();

    // ---- Phase 3: loss terms + fixed-order reduction ----
    float p0 = 0.0f, p1 = 0.0f, p2 = 0.0f;
    if (tid < K_) {
        float d2 = sDmin[tid];
        float s  = scores[((size_t)b * K_ + tid) * N_ + n];
        if (d2 < 1.0f) {                              // dmin < 1  <=>  d2 < 1
            float d = sqrtf(d2);
            p0 = 10000.0f / (1.0f + __expf(d)) * __logf(s);
        } else {
            p1 = __logf(1.0f - s);
            p2 = 1.0f;
        }
    }
    if (tid < 128) { sR0[tid] = p0; sR1[tid] = p1; sR2[tid] = p2; }
    __syncthreads();
    for (int s = 64; s > 0; s >>= 1) {
        if (tid < s) {
            sR0[tid] += sR0[tid + s];
            sR1[tid] += sR1[tid + s];
            sR2[tid] += sR2[tid + s];
        }
        __syncthreads();
    }
    if (tid == 0) {
        partials[bn * 4 + 0] = sR0[0];
        partials[bn * 4 + 1] = sR1[0];
        partials[bn * 4 + 2] = sR2[0];
        partials[bn * 4 + 3] = 0.0f;
    }
}

__global__ __launch_bounds__(128) void keypoint_loss_final(
    const float* __restrict__ partials, float* __restrict__ out)
{
    __shared__ float r0[128], r1[128], r2[128];
    const int tid = threadIdx.x;
    r0[tid] = partials[tid * 4 + 0];
    r1[tid] = partials[tid * 4 + 1];
    r2[tid] = partials[tid * 4 + 2];
    __syncthreads();
    for (int s = 64; s > 0; s >>= 1) {
        if (tid < s) {
            r0[tid] += r0[tid + s];
            r1[tid] += r1[tid + s];
            r2[tid] += r2[tid + s];
        }
        __syncthreads();
    }
    if (tid == 0) {
        float pos = r0[0], neg = r1[0], cnt = r2[0];
        out[0] = -pos - ((cnt > 0.0f) ? (10000.0f / cnt) * neg : 0.0f);
    }
}

extern "C" void kernel_launch(void* const* d_in, const int* in_sizes, int n_in,
                              void* d_out, int out_size, void* d_ws, size_t ws_size,
                              hipStream_t stream) {
    const float* scores = (const float*)d_in[0];  // [B,K,N]
    const float* heat   = (const float*)d_in[1];  // [B,N,H,W]
    const float* kpts   = (const float*)d_in[2];  // [B,K,2]
    float* out      = (float*)d_out;
    float* partials = (float*)d_ws;               // 128*4 floats

    keypoint_loss_main<<<B_ * N_, THREADS, 0, stream>>>(scores, heat, kpts, partials);
    keypoint_loss_final<<<1, 128, 0, stream>>>(partials, out);
}